// LiteMLA_2010044694640
// MI455X (gfx1250) — compile-verified
//
#include <hip/hip_runtime.h>
#include <hip/hip_bf16.h>

typedef __attribute__((ext_vector_type(2))) float v2f;
typedef __attribute__((ext_vector_type(8))) float v8f;

#define DIMB 4
#define NPIX 4096         // 64*64
#define C_IN 512
#define C_QKV 1536
#define C_ATTN 1024
#define GROUPS 192

// ---------------------------------------------------------------------------
// Async global->LDS copies (CDNA5, tracked by ASYNCcnt, bypass VGPRs).
// lds_off: byte offset within workgroup LDS (low 32 bits of generic pointer).
// ---------------------------------------------------------------------------
__device__ __forceinline__ void async_copy_b128(unsigned lds_off, const float* g)
{
    asm volatile("global_load_async_to_lds_b128 %0, %1, off"
                 :: "v"(lds_off), "v"(g)
                 : "memory");
}
__device__ __forceinline__ void async_copy_b32(unsigned lds_off, const float* g)
{
    asm volatile("global_load_async_to_lds_b32 %0, %1, off"
                 :: "v"(lds_off), "v"(g)
                 : "memory");
}
__device__ __forceinline__ void wait_async0()
{
    asm volatile("s_wait_asynccnt 0x0" ::: "memory");
}

// ---------------------------------------------------------------------------
// GEMM: C[b] = A(MxK, row-major) @ X[b](KxN, row-major), optional fused BN.
// Tile: 128(M) x 64(N) per block (256 threads = 8 waves), K-chunk = 32.
// Double-buffered LDS staged by async copies; V_WMMA_F32_16X16X4_F32.
// A tile:  Ws[m][k], stride 36 (144B rows, 16B aligned -> async B128)
// X tile:  Xs[n][k] TRANSPOSED, stride 34 (B frag = contiguous float2/b64;
//          staged by async B32 scatter, global side coalesced)
// ---------------------------------------------------------------------------
#define TM 128
#define TN 64
#define TK 32
#define WSTR 36
#define XSTR 34

__global__ __launch_bounds__(256) void gemm_wmma_f32(
    const float* __restrict__ A, const float* __restrict__ X,
    float* __restrict__ C, int M, int K, int N,
    const float* __restrict__ bn_g, const float* __restrict__ bn_b,
    const float* __restrict__ bn_m, const float* __restrict__ bn_v)
{
    __shared__ float Ws[2][TM * WSTR];   // 2 x 18.4KB
    __shared__ float Xs[2][TN * XSTR];   // 2 x 8.7KB

    const int tid  = threadIdx.x;
    const int lane = tid & 31;
    const int wave = tid >> 5;            // 0..7 -> 16-row M strip
    const int l16  = lane & 15;
    const int kh   = (lane >> 4) << 1;    // 0 for lanes 0-15, 2 for 16-31

    const int b  = blockIdx.z;
    const int m0 = blockIdx.y * TM;
    const int n0 = blockIdx.x * TN;
    const float* Xb = X + (size_t)b * K * N;
    float*       Cb = C + (size_t)b * M * N;

    // staging coordinates
    const int ar  = tid >> 3;             // A row base 0..31 (+i*32)
    const int ac4 = (tid & 7) << 2;       // A col 0,4,...,28
    const int xn  = tid & 63;             // X col (coalesced within wave)
    const int xkb = tid >> 6;             // X row base 0..3 (+i*4)

    const unsigned wbase0 = (unsigned)(size_t)&Ws[0][0];
    const unsigned wbase1 = (unsigned)(size_t)&Ws[1][0];
    const unsigned xbase0 = (unsigned)(size_t)&Xs[0][0];
    const unsigned xbase1 = (unsigned)(size_t)&Xs[1][0];

    auto stage = [&](int bufsel, int k0) {
        const unsigned wb = bufsel ? wbase1 : wbase0;
        const unsigned xb = bufsel ? xbase1 : xbase0;
        #pragma unroll
        for (int i = 0; i < 4; ++i) {
            const int row = ar + i * 32;
            async_copy_b128(wb + (unsigned)(row * WSTR + ac4) * 4u,
                            A + (size_t)(m0 + row) * K + k0 + ac4);
        }
        #pragma unroll
        for (int i = 0; i < 8; ++i) {
            const int kr = xkb + i * 4;   // k row 0..31
            async_copy_b32(xb + (unsigned)(xn * XSTR + kr) * 4u,
                           Xb + (size_t)(k0 + kr) * N + n0 + xn);
        }
    };

    const v8f vz = {0.f,0.f,0.f,0.f,0.f,0.f,0.f,0.f};
    v8f acc[4] = {vz, vz, vz, vz};

    stage(0, 0);
    wait_async0();
    __syncthreads();

    int buf = 0;
    const int mw = wave * 16;
    for (int k0 = 0; k0 < K; k0 += TK) {
        if (k0 + TK < K) stage(buf ^ 1, k0 + TK);   // overlap next-chunk loads

        const float* ws = &Ws[buf][0];
        const float* xs = &Xs[buf][0];
        #pragma unroll
        for (int k = 0; k < TK; k += 4) {
            // A frag 16x4: lanes 0-15 -> (K=k,k+1), lanes 16-31 -> (K=k+2,k+3)
            const v2f af = *(const v2f*)&ws[(mw + l16) * WSTR + k + kh];
            #pragma unroll
            for (int s = 0; s < 4; ++s) {
                // B frag 4x16: contiguous float2 in transposed Xs -> ds_load_b64
                const v2f bf = *(const v2f*)&xs[(s * 16 + l16) * XSTR + k + kh];
                acc[s] = __builtin_amdgcn_wmma_f32_16x16x4_f32(
                    false, af, false, bf, (short)0, acc[s], false, false);
            }
        }
        wait_async0();       // next buffer fully staged
        __syncthreads();     // all waves done reading current buffer
        buf ^= 1;
    }

    // ---- epilogue: D VGPR r => rows (r, r+8), lanes 0-15 / 16-31
    const int rbase = m0 + wave * 16 + ((lane >> 4) << 3);
    #pragma unroll
    for (int s = 0; s < 4; ++s) {
        const int col = n0 + s * 16 + l16;
        #pragma unroll
        for (int r = 0; r < 8; ++r) {
            const int row = rbase + r;
            float v = acc[s][r];
            if (bn_g) {
                const float sc = bn_g[row] * rsqrtf(bn_v[row] + 1e-5f);
                v = v * sc + (bn_b[row] - bn_m[row] * sc);
            }
            Cb[(size_t)row * N + col] = v;
        }
    }
}

// ---------------------------------------------------------------------------
// Fused depthwise 5x5 (pad 2) + grouped 8x8 pointwise.
// ---------------------------------------------------------------------------
__global__ __launch_bounds__(256) void dwpw_kernel(
    const float* __restrict__ qkv, const float* __restrict__ wdw,
    const float* __restrict__ wpw, float* __restrict__ y)
{
    __shared__ float sm[20 * 21];
    const int tid = threadIdx.x;
    const int tx = tid & 15, ty = tid >> 4;
    const int w0 = blockIdx.x * 16, h0 = blockIdx.y * 16;
    const int bg = blockIdx.z;
    const int b = bg / GROUPS, g = bg % GROUPS;

    float acc[8];
    #pragma unroll
    for (int o = 0; o < 8; ++o) acc[o] = 0.f;

    for (int i = 0; i < 8; ++i) {
        const int gc = g * 8 + i;
        const float* plane = qkv + ((size_t)b * C_QKV + gc) * NPIX;
        for (int p = tid; p < 400; p += 256) {
            const int rr = p / 20, cc = p % 20;
            const int gh = h0 + rr - 2, gw = w0 + cc - 2;
            float v = 0.f;
            if (gh >= 0 && gh < 64 && gw >= 0 && gw < 64)
                v = plane[gh * 64 + gw];
            sm[rr * 21 + cc] = v;
        }
        __syncthreads();

        const float* wd = wdw + (size_t)gc * 25;
        float dwv = 0.f;
        #pragma unroll
        for (int dy = 0; dy < 5; ++dy)
            #pragma unroll
            for (int dx = 0; dx < 5; ++dx)
                dwv = fmaf(wd[dy * 5 + dx], sm[(ty + dy) * 21 + (tx + dx)], dwv);

        const float* wp = wpw + (size_t)g * 64 + i;   // wpw[g][o][i]
        #pragma unroll
        for (int o = 0; o < 8; ++o)
            acc[o] = fmaf(wp[o * 8], dwv, acc[o]);
        __syncthreads();
    }

    const int n = (h0 + ty) * 64 + (w0 + tx);
    #pragma unroll
    for (int o = 0; o < 8; ++o)
        y[((size_t)b * C_QKV + g * 8 + o) * NPIX + n] = acc[o];
}

// ---------------------------------------------------------------------------
// Attention pass A: vk[b,h,d,e] = sum_n v_d(n) * relu(k_e(n)),  d=0..8 (d=8: ones)
// ---------------------------------------------------------------------------
__global__ __launch_bounds__(256) void attn_vk_kernel(
    const float* __restrict__ qkv, const float* __restrict__ y,
    float* __restrict__ vk)
{
    __shared__ float red[8][72];
    const int bh = blockIdx.x;
    const int b = bh >> 7, h = bh & 127;
    const int tid = threadIdx.x, lane = tid & 31, wave = tid >> 5;

    const float* base = (h < 64)
        ? qkv + ((size_t)b * C_QKV + h * 24) * NPIX
        : y   + ((size_t)b * C_QKV + (h - 64) * 24) * NPIX;
    const float* kch = base + 8 * NPIX;
    const float* vch = base + 16 * NPIX;

    float acc[72];
    #pragma unroll
    for (int i = 0; i < 72; ++i) acc[i] = 0.f;

    for (int n = tid; n < NPIX; n += 256) {
        float kv[8], vv[8];
        #pragma unroll
        for (int e = 0; e < 8; ++e) kv[e] = fmaxf(kch[e * NPIX + n], 0.f);
        #pragma unroll
        for (int d = 0; d < 8; ++d) vv[d] = vch[d * NPIX + n];
        #pragma unroll
        for (int d = 0; d < 8; ++d)
            #pragma unroll
            for (int e = 0; e < 8; ++e)
                acc[d * 8 + e] = fmaf(vv[d], kv[e], acc[d * 8 + e]);
        #pragma unroll
        for (int e = 0; e < 8; ++e) acc[64 + e] += kv[e];   // ones row
    }

    #pragma unroll
    for (int i = 0; i < 72; ++i) {
        float v = acc[i];
        for (int off = 16; off > 0; off >>= 1) v += __shfl_down(v, off, 32);
        if (lane == 0) red[wave][i] = v;
    }
    __syncthreads();
    if (tid < 72) {
        float s = 0.f;
        #pragma unroll
        for (int w = 0; w < 8; ++w) s += red[w][tid];
        vk[(size_t)bh * 72 + tid] = s;
    }
}

// ---------------------------------------------------------------------------
// Attention pass B: out[b,h*8+d,n] = (vk[d,:].q(n)) / (vk[8,:].q(n) + eps)
// ---------------------------------------------------------------------------
__global__ __launch_bounds__(256) void attn_apply_kernel(
    const float* __restrict__ qkv, const float* __restrict__ y,
    const float* __restrict__ vk, float* __restrict__ out)
{
    __shared__ float svk[72];
    const int bh = blockIdx.y;
    const int b = bh >> 7, h = bh & 127;
    const int tid = threadIdx.x;
    if (tid < 72) svk[tid] = vk[(size_t)bh * 72 + tid];
    __syncthreads();

    const float* base = (h < 64)
        ? qkv + ((size_t)b * C_QKV + h * 24) * NPIX
        : y   + ((size_t)b * C_QKV + (h - 64) * 24) * NPIX;

    const int n = blockIdx.x * 256 + tid;
    float q[8];
    #pragma unroll
    for (int e = 0; e < 8; ++e) q[e] = fmaxf(base[e * NPIX + n], 0.f);

    float den = 0.f;
    #pragma unroll
    for (int e = 0; e < 8; ++e) den = fmaf(svk[64 + e], q[e], den);
    const float inv = 1.f / (den + 1e-15f);

    #pragma unroll
    for (int d = 0; d < 8; ++d) {
        float num = 0.f;
        #pragma unroll
        for (int e = 0; e < 8; ++e) num = fmaf(svk[d * 8 + e], q[e], num);
        out[((size_t)b * C_ATTN + h * 8 + d) * NPIX + n] = num * inv;
    }
}

// ---------------------------------------------------------------------------
extern "C" void kernel_launch(void* const* d_in, const int* in_sizes, int n_in,
                              void* d_out, int out_size, void* d_ws, size_t ws_size,
                              hipStream_t stream)
{
    const float* x      = (const float*)d_in[0];
    const float* w_qkv  = (const float*)d_in[1];
    const float* w_dw   = (const float*)d_in[2];
    const float* w_pw   = (const float*)d_in[3];
    const float* w_proj = (const float*)d_in[4];
    const float* bn_g   = (const float*)d_in[5];
    const float* bn_b   = (const float*)d_in[6];
    const float* bn_m   = (const float*)d_in[7];
    const float* bn_v   = (const float*)d_in[8];
    float* out = (float*)d_out;

    const size_t QKV_BYTES  = (size_t)DIMB * C_QKV * NPIX * sizeof(float);   // 100.7 MB
    const size_t ATTN_BYTES = (size_t)DIMB * C_ATTN * NPIX * sizeof(float);  // 67.1 MB
    char* ws = (char*)d_ws;
    float* qkv  = (float*)(ws);
    float* ybuf = (float*)(ws + QKV_BYTES);
    float* attn = (float*)(ws + 2 * QKV_BYTES);
    float* vk   = (float*)(ws + 2 * QKV_BYTES + ATTN_BYTES);

    // 1) qkv = w_qkv @ x          (M=1536, K=512, N=4096 per batch)
    gemm_wmma_f32<<<dim3(NPIX / TN, C_QKV / TM, DIMB), 256, 0, stream>>>(
        w_qkv, x, qkv, C_QKV, C_IN, NPIX, nullptr, nullptr, nullptr, nullptr);

    // 2) y = grouped-pointwise(depthwise5x5(qkv))
    dwpw_kernel<<<dim3(4, 4, DIMB * GROUPS), 256, 0, stream>>>(qkv, w_dw, w_pw, ybuf);

    // 3a) vk accumulation per (b, head)
    attn_vk_kernel<<<DIMB * 128, 256, 0, stream>>>(qkv, ybuf, vk);

    // 3b) normalized attention output (B, 1024, N)
    attn_apply_kernel<<<dim3(NPIX / 256, DIMB * 128), 256, 0, stream>>>(qkv, ybuf, vk, attn);

    // 4) out = BN(w_proj @ attn)  (M=512, K=1024, N=4096 per batch, fused BN)
    gemm_wmma_f32<<<dim3(NPIX / TN, C_IN / TM, DIMB), 256, 0, stream>>>(
        w_proj, attn, out, C_IN, C_ATTN, NPIX, bn_g, bn_b, bn_m, bn_v);
}